// YOLOv1Loss2D_46084999086167
// MI455X (gfx1250) — compile-verified
//
#include <hip/hip_runtime.h>
#include <hip/hip_bf16.h>
#include <cstdint>

// ---------------------------------------------------------------------------
// YOLOv1 loss on MI455X (gfx1250).
// Memory-bound streaming reduction: 193 MB read @ 23.3 TB/s => ~8.3us floor.
// CDNA5 path: DOUBLE-BUFFERED async global->LDS copies (ASYNCcnt pipeline),
// global_prefetch of tile+2, LDS-staged conflict-free compute, wave32-safe
// reductions. No matmul structure => WMMA inapplicable.
// ---------------------------------------------------------------------------

#define DCOLS       30
#define TILE_ROWS   128
#define THREADS     128
#define TILE_FLOATS (TILE_ROWS * DCOLS)          // 3840 floats
#define TILE_BYTES  (TILE_FLOATS * 4)            // 15360 bytes
#define SWEEPS      (TILE_BYTES / (THREADS * 8)) // 15 x b64 per thread per array
#define OPS_PER_TILE (2 * SWEEPS)                // 30 async instrs per tile issue

// --- CDNA5 async global->LDS copy (tracked by ASYNCcnt) --------------------
//   void __builtin_amdgcn_global_load_async_to_lds_b64(
//       v2i addrspace(1)* src, v2i addrspace(3)* dst, imm offset, imm cpol)
typedef int v2i_t __attribute__((vector_size(8)));

#if __has_builtin(__builtin_amdgcn_global_load_async_to_lds_b64)
#define ASYNC_LDS_B64(lds, g)                                                  \
  __builtin_amdgcn_global_load_async_to_lds_b64(                               \
      (__attribute__((address_space(1))) v2i_t*)(uintptr_t)(g),                \
      (__attribute__((address_space(3))) v2i_t*)(uintptr_t)(lds), 0, 0)
#else
#define ASYNC_LDS_B64(lds, g)                                                  \
  asm volatile("global_load_async_to_lds_b64 %0, %1, off"                      \
               :: "v"((unsigned)(uintptr_t)(lds)),                             \
                  "v"((unsigned long long)(uintptr_t)(g))                      \
               : "memory")
#endif

#if __has_builtin(__builtin_amdgcn_s_wait_asynccnt)
#define WAIT_ASYNC(n) __builtin_amdgcn_s_wait_asynccnt(n)
#else
#define WAIT_ASYNC(n) asm volatile("s_wait_asynccnt %0" :: "i"(n) : "memory")
#endif

// --- per-cell loss math -----------------------------------------------------
__device__ __forceinline__ float iou_yolo(float ax, float ay, float aw, float ah,
                                          float bx, float by, float bw, float bh) {
  float ax1 = ax - aw * 0.5f, ay1 = ay - ah * 0.5f;
  float ax2 = ax + aw * 0.5f, ay2 = ay + ah * 0.5f;
  float bx1 = bx - bw * 0.5f, by1 = by - bh * 0.5f;
  float bx2 = bx + bw * 0.5f, by2 = by + bh * 0.5f;
  float iw = fmaxf(fminf(ax2, bx2) - fmaxf(ax1, bx1), 0.0f);
  float ih = fmaxf(fminf(ay2, by2) - fmaxf(ay1, by1), 0.0f);
  float inter  = iw * ih;
  float area_a = fabsf((ax2 - ax1) * (ay2 - ay1));
  float area_b = fabsf((bx2 - bx1) * (by2 - by1));
  return inter / (area_a + area_b - inter + 1e-6f);
}

__device__ __forceinline__ float row_term(const float* __restrict__ tr,
                                          const float* __restrict__ pr) {
  float b0 = tr[0], b1 = tr[1], b2 = tr[2], b3 = tr[3], t4 = tr[4];
  float p0 = pr[0], p1 = pr[1], p2 = pr[2], p3 = pr[3], c1 = pr[4];
  float q0 = pr[5], q1 = pr[6], q2 = pr[7], q3 = pr[8], c2 = pr[9];

  float i1 = iou_yolo(b0, b1, b2, b3, p0, p1, p2, p3);
  float i2 = iou_yolo(b0, b1, b2, b3, q0, q1, q2, q3);
  bool use1 = i1 > i2;

  float bh0 = use1 ? p0 : q0, bh1 = use1 ? p1 : q1;
  float bh2 = use1 ? p2 : q2, bh3 = use1 ? p3 : q3;
  float cc  = use1 ? c1 : c2;   // chosen confidence
  float co  = use1 ? c2 : c1;   // other confidence

  float dx = b0 - bh0, dy = b1 - bh1;
  float xy = 5.0f * (dx * dx + dy * dy);

  float sw = sqrtf(b2) - sqrtf(fabsf(bh2 + 1e-6f));
  float sh = sqrtf(b3) - sqrtf(fabsf(bh3 + 1e-6f));
  float wh = 5.0f * (sw * sw + sh * sh);

  float dc = t4 - cc;
  float obj = xy + wh + dc * dc + 0.5f * co * co;

  float cls = 0.0f;
#pragma unroll
  for (int k = 10; k < DCOLS; ++k) {
    float d = tr[k] - pr[k];
    cls = fmaf(d, d, cls);
  }
  obj += cls;

  float d4 = t4 - c1, d9 = t4 - c2;
  float noobj = 0.5f * (d4 * d4 + d9 * d9);

  return (t4 == 1.0f) ? obj : noobj;
}

// --- stage 1: per-block partial sums (double-buffered async pipeline) ------
__global__ __launch_bounds__(THREADS)
void yolo_loss_main(const float* __restrict__ t, const float* __restrict__ p,
                    int rows, float* __restrict__ partials) {
  // 2 buffers x (trues + preds) = 4 * 15360 B = 61440 B + 512 B reduce < 64 KB
  __shared__ float lds_t[2][TILE_FLOATS];
  __shared__ float lds_p[2][TILE_FLOATS];
  __shared__ float red[THREADS];

  const int tid = threadIdx.x;
  const int G   = gridDim.x;
  const int ntiles_full = rows / TILE_ROWS;

  float acc = 0.0f;

  // Prologue: start streaming this block's first tile into buffer 0.
  int tile0 = blockIdx.x;
  if (tile0 < ntiles_full) {
    const char* gt = (const char*)t + (size_t)tile0 * TILE_BYTES;
    const char* gp = (const char*)p + (size_t)tile0 * TILE_BYTES;
#pragma unroll
    for (int k = 0; k < SWEEPS; ++k) {
      unsigned off = (unsigned)(tid * 8 + k * (THREADS * 8));
      ASYNC_LDS_B64((char*)lds_t[0] + off, gt + off);
      ASYNC_LDS_B64((char*)lds_p[0] + off, gp + off);
    }
  }

  int buf = 0;
  for (int tile = tile0; tile < ntiles_full; tile += G, buf ^= 1) {
    // Warm L2 two tiles ahead of the async engine.
    int pf = tile + 2 * G;
    if (pf < ntiles_full) {
      size_t poff = (size_t)tid * 128;   // 128 thr * 128 B = 16 KB >= tile
      if (poff < TILE_BYTES) {
        __builtin_prefetch((const char*)t + (size_t)pf * TILE_BYTES + poff, 0, 0);
        __builtin_prefetch((const char*)p + (size_t)pf * TILE_BYTES + poff, 0, 0);
      }
    }

    // Issue next tile's async loads into the other buffer, then wait for the
    // CURRENT tile: async ops retire in order, so counter<=OPS_PER_TILE means
    // everything issued before the just-queued batch has landed in LDS.
    int next = tile + G;
    if (next < ntiles_full) {
      const char* gt = (const char*)t + (size_t)next * TILE_BYTES;
      const char* gp = (const char*)p + (size_t)next * TILE_BYTES;
#pragma unroll
      for (int k = 0; k < SWEEPS; ++k) {
        unsigned off = (unsigned)(tid * 8 + k * (THREADS * 8));
        ASYNC_LDS_B64((char*)lds_t[buf ^ 1] + off, gt + off);
        ASYNC_LDS_B64((char*)lds_p[buf ^ 1] + off, gp + off);
      }
      WAIT_ASYNC(OPS_PER_TILE);
    } else {
      WAIT_ASYNC(0);
    }
    __syncthreads();   // current tile visible to all waves

    // One row per thread; stride-30 dwords across wave32 lanes is
    // bank-conflict-free (gcd(30,64)=2 => 32 distinct banks).
    acc += row_term(&lds_t[buf][tid * DCOLS], &lds_p[buf][tid * DCOLS]);

    __syncthreads();   // protect buffer before next iteration's async writes
  }

  // Tail rows (rows % TILE_ROWS), handled once by the owning block.
  if (ntiles_full * TILE_ROWS < rows &&
      blockIdx.x == (unsigned)(ntiles_full % G)) {
    int row = ntiles_full * TILE_ROWS + tid;
    if (row < rows)
      acc += row_term(t + (size_t)row * DCOLS, p + (size_t)row * DCOLS);
  }

  // Block reduction (wave32-agnostic LDS tree).
  red[tid] = acc;
  __syncthreads();
#pragma unroll
  for (int s = THREADS / 2; s > 0; s >>= 1) {
    if (tid < s) red[tid] += red[tid + s];
    __syncthreads();
  }
  if (tid == 0) partials[blockIdx.x] = red[0];
}

// --- stage 2: reduce partials, scale by 1/batch ----------------------------
#define RTHREADS 256
__global__ __launch_bounds__(RTHREADS)
void yolo_loss_reduce(const float* __restrict__ partials, int n,
                      float* __restrict__ out, float scale) {
  __shared__ float red[RTHREADS];
  float a = 0.0f;
  for (int i = threadIdx.x; i < n; i += RTHREADS) a += partials[i];
  red[threadIdx.x] = a;
  __syncthreads();
#pragma unroll
  for (int s = RTHREADS / 2; s > 0; s >>= 1) {
    if (threadIdx.x < s) red[threadIdx.x] += red[threadIdx.x + s];
    __syncthreads();
  }
  if (threadIdx.x == 0) out[0] = red[0] * scale;
}

// ---------------------------------------------------------------------------
extern "C" void kernel_launch(void* const* d_in, const int* in_sizes, int n_in,
                              void* d_out, int out_size, void* d_ws, size_t ws_size,
                              hipStream_t stream) {
  (void)n_in; (void)out_size; (void)ws_size;
  const float* y_trues = (const float*)d_in[0];
  const float* y_preds = (const float*)d_in[1];

  const int total  = in_sizes[0];        // BATCH*S*S*D
  const int rows   = total / DCOLS;      // BATCH*S*S cells
  const int ntiles = (rows + TILE_ROWS - 1) / TILE_ROWS;
  const int grid   = ntiles < 2048 ? ntiles : 2048;

  float* partials = (float*)d_ws;        // grid floats of scratch

  yolo_loss_main<<<grid, THREADS, 0, stream>>>(y_trues, y_preds, rows, partials);

  // total / batch_size, batch = rows / (S*S) = rows / 49
  const float scale = 49.0f / (float)rows;
  yolo_loss_reduce<<<1, RTHREADS, 0, stream>>>(partials, grid, (float*)d_out, scale);
}